// FraudDetectionHybrid_65481071395227
// MI455X (gfx1250) — compile-verified
//
#include <hip/hip_runtime.h>

typedef __attribute__((ext_vector_type(2))) float v2f;
typedef __attribute__((ext_vector_type(8))) float v8f;

#define RSIZE 4096
#define BATCH 65536
#define LOG2E 1.44269504088896340736f

__device__ __forceinline__ float fast_tanh(float z) {
    // tanh(z) = 1 - 2/(exp(2z)+1), native v_exp_f32 + v_rcp_f32
    float t = __builtin_amdgcn_exp2f(2.0f * LOG2E * z);
    return 1.0f - 2.0f * __builtin_amdgcn_rcpf(t + 1.0f);
}

__global__ __launch_bounds__(256) void fraud_fused(
    const float* __restrict__ x,     const float* __restrict__ w0,
    const float* __restrict__ b0,    const float* __restrict__ wl,
    const float* __restrict__ bl,    const float* __restrict__ w_out,
    const float* __restrict__ b_out, const float* __restrict__ ref,
    const float* __restrict__ w_cls, const float* __restrict__ b_cls,
    float* __restrict__ out)
{
    // Interleaved (sy_j, c_j) pairs: 4096 * 8B = 32 KB of the 320 KB WGP LDS
    __shared__ __align__(16) float s_syc[2 * RSIZE];

    // ---- prep: per-reference constants, folded classifier weight ----
    for (int j = (int)threadIdx.x; j < RSIZE; j += 256) {
        float y0  = ref[2 * j + 0];
        float y1  = ref[2 * j + 1];
        float sy  = y0 + y1;
        float sy2 = y0 * y0 + y1 * y1;
        float c   = w_cls[j] * __builtin_amdgcn_exp2f(-LOG2E * sy2);
        s_syc[2 * j + 0] = sy;
        s_syc[2 * j + 1] = c;
    }
    __syncthreads();

    int wave = threadIdx.x >> 5;
    int lane = threadIdx.x & 31;
    int tile = blockIdx.x * 8 + wave;   // 16 samples per wave-tile
    int m    = lane & 15;
    int sample = tile * 16 + m;

    // ---- backbone: scalar feature f (lanes L and L+16 redundantly compute row M=L) ----
    float x0 = x[2 * sample + 0];
    float x1 = x[2 * sample + 1];
    float h0 = fast_tanh(fmaf(x0, w0[0], fmaf(x1, w0[1], b0[0]))) * 0.1f;
    float h1 = fast_tanh(fmaf(x0, w0[2], fmaf(x1, w0[3], b0[1]))) * 0.1f;
#pragma unroll
    for (int i = 0; i < 3; ++i) {
        float t0 = fast_tanh(fmaf(h0, wl[4*i+0], fmaf(h1, wl[4*i+1], bl[2*i+0]))) * 0.05f;
        float t1 = fast_tanh(fmaf(h0, wl[4*i+2], fmaf(h1, wl[4*i+3], bl[2*i+1]))) * 0.05f;
        h0 = t0; h1 = t1;
    }
    float f = fmaf(h0, w_out[0], fmaf(h1, w_out[1], b_out[0]));

    // kernel factorization: c_j * exp2(p*sy_j + q)
    float p = 2.0f * f * LOG2E;
    float q = -2.0f * f * f * LOG2E;

    const float4* s4 = reinterpret_cast<const float4*>(s_syc);
    int khalfpair = (lane >> 4);  // lanes 0-15: K=0,1 ; lanes 16-31: K=2,3

    v8f acc = {0.f,0.f,0.f,0.f,0.f,0.f,0.f,0.f};
#pragma unroll 4
    for (int j0 = 0; j0 < RSIZE; j0 += 4) {
        // one ds_load_b128: (sy_j, c_j, sy_{j+1}, c_{j+1}) with j = j0 + 2*khalfpair
        float4 d = s4[(j0 >> 1) + khalfpair];
        v2f a, b;
        a.x = __builtin_amdgcn_exp2f(fmaf(p, d.x, q));  // A[m, K]
        a.y = __builtin_amdgcn_exp2f(fmaf(p, d.z, q));  // A[m, K+1]
        b.x = d.y;                                       // B[K,   n] (replicated over n)
        b.y = d.w;                                       // B[K+1, n]
        // D = A(16x4 exps) x B(4x16 c-broadcast) + C : reduction on the matrix pipe
        acc = __builtin_amdgcn_wmma_f32_16x16x4_f32(
            false, a, false, b, (short)0, acc, false, false);
    }

    // D[m, n] identical across n; C/D layout: VGPR v = row v (lanes 0-15) / row v+8 (lanes 16-31)
    float bc = b_cls[0];
    if (lane == 0) {
#pragma unroll
        for (int v = 0; v < 8; ++v) out[tile * 16 + v] = acc[v] + bc;
    } else if (lane == 16) {
#pragma unroll
        for (int v = 0; v < 8; ++v) out[tile * 16 + 8 + v] = acc[v] + bc;
    }
}

extern "C" void kernel_launch(void* const* d_in, const int* in_sizes, int n_in,
                              void* d_out, int out_size, void* d_ws, size_t ws_size,
                              hipStream_t stream)
{
    const float* x     = (const float*)d_in[0];
    const float* w0    = (const float*)d_in[1];
    const float* b0    = (const float*)d_in[2];
    const float* wl    = (const float*)d_in[3];
    const float* bl    = (const float*)d_in[4];
    const float* w_out = (const float*)d_in[5];
    const float* b_out = (const float*)d_in[6];
    const float* ref   = (const float*)d_in[7];
    const float* w_cls = (const float*)d_in[8];
    const float* b_cls = (const float*)d_in[9];
    float* out = (float*)d_out;

    dim3 grid(BATCH / (16 * 8));  // 512 blocks, 8 wave-tiles each
    dim3 block(256);
    hipLaunchKernelGGL(fraud_fused, grid, block, 0, stream,
                       x, w0, b0, wl, bl, w_out, b_out, ref, w_cls, b_cls, out);
}